// RUNG_learnable_gamma_22247930593891
// MI455X (gfx1250) — compile-verified
//
#include <hip/hip_runtime.h>
#include <math.h>

typedef float v2f __attribute__((ext_vector_type(2)));
typedef float v4f __attribute__((ext_vector_type(4)));
typedef float v8f __attribute__((ext_vector_type(8)));
typedef int   v4i_t __attribute__((__vector_size__(16)));

#define IN_DIM 512
#define HID    256
#define OUT_C  16
#define K_STEPS 10
#define SCAD_A_C 3.7f
#define EPS_C 1e-8f
// LAM = 1/0.9 - 1
#define LAM_C 0.11111111111111116f
#define WPB 8   // waves per block in the hot kernel

#if defined(__has_builtin)
#if __has_builtin(__builtin_amdgcn_global_load_async_to_lds_b128)
#define RUNG_HAVE_ASYNC 1
#endif
#endif

#if defined(RUNG_HAVE_ASYNC)
// Signature (from hipcc diagnostic): params are generic pointers to int4.
//   (global_src, lds_dst, imm offset, imm cpol)
#define G2L_B128(gp, lp)                                                       \
  __builtin_amdgcn_global_load_async_to_lds_b128((v4i_t*)(gp), (v4i_t*)(lp),   \
                                                 0, 0)
#endif

static __device__ __forceinline__ void rung_wait_async() {
#if defined(RUNG_HAVE_ASYNC)
#if __has_builtin(__builtin_amdgcn_s_wait_asynccnt)
  __builtin_amdgcn_s_wait_asynccnt(0);
#else
  asm volatile("s_wait_asynccnt 0x0" ::: "memory");
#endif
#endif
}

static __device__ __forceinline__ v2f mk2(float a, float b) {
  v2f r; r[0] = a; r[1] = b; return r;
}

// ---------------------------------------------------------------------------
// MLP embedding: F0 = relu(F @ W1 + b1) @ W2 + b2   (one block per row)
// ---------------------------------------------------------------------------
__global__ __launch_bounds__(256) void rung_mlp_kernel(
    const float* __restrict__ F, const float* __restrict__ W1,
    const float* __restrict__ b1, const float* __restrict__ W2,
    const float* __restrict__ b2, float* __restrict__ F0, int n) {
  __shared__ float frow[IN_DIM];
  __shared__ float hrow[HID];
  __shared__ float pred[256];
  const int i = blockIdx.x;
  const int t = threadIdx.x;
  for (int k = t; k < IN_DIM; k += 256) frow[k] = F[(size_t)i * IN_DIM + k];
  __syncthreads();
  float acc = b1[t];
  #pragma unroll 8
  for (int k = 0; k < IN_DIM; ++k) acc = fmaf(frow[k], W1[(size_t)k * HID + t], acc);
  hrow[t] = fmaxf(acc, 0.0f);
  __syncthreads();
  const int c = t & 15, seg = t >> 4;
  float p = 0.0f;
  #pragma unroll
  for (int u = 0; u < 16; ++u) {
    const int hh = seg * 16 + u;
    p = fmaf(hrow[hh], W2[hh * OUT_C + c], p);
  }
  pred[t] = p;
  __syncthreads();
  if (t < OUT_C) {
    float s = b2[t];
    #pragma unroll
    for (int g = 0; g < 16; ++g) s += pred[g * 16 + t];
    F0[(size_t)i * OUT_C + t] = s;
  }
}

// ---------------------------------------------------------------------------
// Degrees with self loops: D = 1 + rowsum(A); Dsq = sqrt(D); Dinv = 1/sqrt(D)
// ---------------------------------------------------------------------------
__global__ __launch_bounds__(256) void rung_degree_kernel(
    const float* __restrict__ A, float* __restrict__ Dd,
    float* __restrict__ Dsq, float* __restrict__ Dinv, int n) {
  __shared__ float red[256];
  const int i = blockIdx.x;
  float s = 0.0f;
  for (int j = threadIdx.x; j < n; j += 256) s += A[(size_t)i * n + j];
  red[threadIdx.x] = s;
  __syncthreads();
  for (int off = 128; off > 0; off >>= 1) {
    if (threadIdx.x < off) red[threadIdx.x] += red[threadIdx.x + off];
    __syncthreads();
  }
  if (threadIdx.x == 0) {
    const float d = red[0] + 1.0f;
    Dd[i] = d;
    const float ds = sqrtf(d);
    Dsq[i] = ds;
    Dinv[i] = 1.0f / ds;
  }
}

// ---------------------------------------------------------------------------
// Per-step prep: Fn = Fk / Dsq; sq = rownorm2(Fn); zero the staging buffers
// ---------------------------------------------------------------------------
__global__ __launch_bounds__(256) void rung_prep_kernel(
    const float* __restrict__ Fk, const float* __restrict__ Dsq,
    float* __restrict__ Fn, float* __restrict__ sq,
    float* __restrict__ numer, float* __restrict__ qsum, int n) {
  const int i = blockIdx.x * blockDim.x + threadIdx.x;
  if (i >= n) return;
  const float inv = 1.0f / Dsq[i];
  float s = 0.0f;
  #pragma unroll
  for (int c = 0; c < OUT_C; ++c) {
    const float v = Fk[i * OUT_C + c] * inv;
    Fn[i * OUT_C + c] = v;
    s = fmaf(v, v, s);
    numer[i * OUT_C + c] = 0.0f;
  }
  sq[i] = s;
  qsum[i] = 0.0f;
}

// ---------------------------------------------------------------------------
// Stage one shared j-tile bundle (Fn 1KB, Fk 1KB, sq/Dinv 128B) into LDS.
// Async path: wave 0 issues GLOBAL_LOAD_ASYNC_TO_LDS_B128 (ASYNCcnt-tracked).
// Fallback: cooperative ds_store staging by all 256 threads.
// ---------------------------------------------------------------------------
static __device__ __forceinline__ void rung_stage_tile(
    const float* __restrict__ Fn, const float* __restrict__ Fk,
    const float* __restrict__ sq, const float* __restrict__ Dinv,
    float* lFn, float* lFk, float* lSD, int j0, int t, int lane, int wave) {
#if defined(RUNG_HAVE_ASYNC)
  if (wave == 0) {
    const float* gFn = Fn + (size_t)j0 * OUT_C;
    const float* gFk = Fk + (size_t)j0 * OUT_C;
    G2L_B128(gFn + lane * 4,       lFn + lane * 4);
    G2L_B128(gFn + 128 + lane * 4, lFn + 128 + lane * 4);
    G2L_B128(gFk + lane * 4,       lFk + lane * 4);
    G2L_B128(gFk + 128 + lane * 4, lFk + 128 + lane * 4);
    if (lane < 8) {
      const float* gsd = (lane < 4) ? (sq + j0 + lane * 4)
                                    : (Dinv + j0 + (lane - 4) * 4);
      float* lsd = (lane < 4) ? (lSD + lane * 4) : (lSD + 16 + (lane - 4) * 4);
      G2L_B128(gsd, lsd);
    }
  }
#else
  lFn[t] = Fn[(size_t)j0 * OUT_C + t];
  lFk[t] = Fk[(size_t)j0 * OUT_C + t];
  if (t < 16) { lSD[t] = sq[j0 + t]; lSD[16 + t] = Dinv[j0 + t]; }
#endif
}

// ---------------------------------------------------------------------------
// Hot kernel: 8 waves/block, wave w owns i-tile (blockIdx.x*8+w); the block
// sweeps a chunk of j-tiles with double-buffered async LDS staging.
//   Gt = Fn_j @ Fn_i^T          (4x v_wmma_f32_16x16x4_f32)
//   W  = scad(sqrt(dist2)) masked by A, degree-normalized
//   numer += W @ Fk_j           (4x v_wmma_f32_16x16x4_f32, shfl-transposed A)
// ---------------------------------------------------------------------------
__global__ __launch_bounds__(256) void rung_step_kernel(
    const float* __restrict__ A, const float* __restrict__ Fn,
    const float* __restrict__ Fk, const float* __restrict__ sq,
    const float* __restrict__ Dinv, const float* __restrict__ log_lams,
    float* __restrict__ numer, float* __restrict__ qsum,
    int k, int n, int tilesPerChunk) {
  __shared__ __align__(16) float ldsFn[2][256];
  __shared__ __align__(16) float ldsFk[2][256];
  __shared__ __align__(16) float ldsSD[2][32];  // [0..15]=sq, [16..31]=Dinv

  const int t = threadIdx.x;
  const int lane = t & 31;
  const int wave = t >> 5;
  const int r = lane & 15;
  const int h = lane >> 4;
  const int i0 = (blockIdx.x * WPB + wave) * 16;
  const int jt0 = blockIdx.y * tilesPerChunk;

  const float lam = expf(log_lams[k]);
  const float alam = SCAD_A_C * lam;
  const float c2 = (SCAD_A_C - 1.0f) * lam * 2.0f;

  // B operand for the Gram matmul: Fn i-tile (resident for the whole sweep)
  v2f bI[4];
  #pragma unroll
  for (int s = 0; s < 4; ++s)
    bI[s] = *(const v2f*)&Fn[(i0 + r) * OUT_C + 4 * s + 2 * h];

  const int gi = i0 + r;
  const float sqi = sq[gi];
  const float dinv_i = Dinv[gi];
  const size_t arow = (size_t)gi * n;

  v8f acc = {};      // numer accumulator, C/D layout: [m = v+8h][c = r]
  float qacc = 0.0f; // partial sum_j W*A for row gi (this half's j's)

  // Prologue: stage first j-tile into buffer 0
  rung_stage_tile(Fn, Fk, sq, Dinv, &ldsFn[0][0], &ldsFk[0][0], &ldsSD[0][0],
                  jt0 * 16, t, lane, wave);

  for (int it = 0; it < tilesPerChunk; ++it) {
    const int cur = it & 1;
    const int j0 = (jt0 + it) * 16;

    if (wave == 0) rung_wait_async();   // s_wait_asynccnt 0: buf[cur] landed
    __syncthreads();                    // visible to all; prev readers done

    if (it + 1 < tilesPerChunk)
      rung_stage_tile(Fn, Fk, sq, Dinv, &ldsFn[cur ^ 1][0], &ldsFk[cur ^ 1][0],
                      &ldsSD[cur ^ 1][0], (jt0 + it + 1) * 16, t, lane, wave);

    const int jb = j0 + 8 * h;
    __builtin_prefetch(&A[arow + j0 + 16], 0, 1);

    // A operand: Fn j-tile from LDS
    v2f aJ[4];
    #pragma unroll
    for (int s = 0; s < 4; ++s)
      aJ[s] = *(const v2f*)&ldsFn[cur][r * OUT_C + 4 * s + 2 * h];

    // Gram tile: Gt[m][nn] = sum_k Fn[j0+m][k] * Fn[i0+nn][k]
    v8f g = {};
    #pragma unroll
    for (int s = 0; s < 4; ++s)
      g = __builtin_amdgcn_wmma_f32_16x16x4_f32(false, aJ[s], false, bI[s],
                                                (short)0, g, false, false);

    // Per-element SCAD weighting; lane holds (i = gi, j = jb + v)
    const v4f a01 = *(const v4f*)&A[arow + jb];
    const v4f a23 = *(const v4f*)&A[arow + jb + 4];
    const v4f sj0 = *(const v4f*)&ldsSD[cur][8 * h];
    const v4f sj1 = *(const v4f*)&ldsSD[cur][8 * h + 4];
    const v4f dj0 = *(const v4f*)&ldsSD[cur][16 + 8 * h];
    const v4f dj1 = *(const v4f*)&ldsSD[cur][16 + 8 * h + 4];

    float wt[8];
    #pragma unroll
    for (int v = 0; v < 8; ++v) {
      const float sqj = (v < 4) ? sj0[v] : sj1[v - 4];
      const float av  = (v < 4) ? a01[v] : a23[v - 4];
      const float djv = (v < 4) ? dj0[v] : dj1[v - 4];
      float Z = fmaxf(sqi + sqj - 2.0f * g[v], 0.0f);
      const float y = sqrtf(Z);
      const float ys = fmaxf(y, EPS_C);
      const float r1 = 0.5f / ys;
      const float d2 = fmaxf(c2 * ys, EPS_C);
      const float r2 = fmaxf((alam - y) / d2, 0.0f);
      float Wv = (y < lam) ? r1 : ((y < alam) ? r2 : 0.0f);
      if (gi == jb + v) Wv = 0.0f;     // zero diagonal
      if (Wv != Wv) Wv = 1.0f;         // NaN -> 1
      const float wa = Wv * av;        // W * A_loop (diag is 0 anyway)
      qacc += wa;
      wt[v] = wa * dinv_i * djv;       // W * A_tilde
    }

    // Cross-half exchange: C/D-layout W tile -> A operand layout
    float pwt[8];
    #pragma unroll
    for (int v = 0; v < 8; ++v) pwt[v] = __shfl_xor(wt[v], 16, 32);

    v2f aw[4];
    aw[0] = (h == 0) ? mk2(wt[0], wt[1]) : mk2(pwt[2], pwt[3]);  // K 0..3
    aw[1] = (h == 0) ? mk2(wt[4], wt[5]) : mk2(pwt[6], pwt[7]);  // K 4..7
    aw[2] = (h == 0) ? mk2(pwt[0], pwt[1]) : mk2(wt[2], wt[3]);  // K 8..11
    aw[3] = (h == 0) ? mk2(pwt[4], pwt[5]) : mk2(wt[6], wt[7]);  // K 12..15

    // numer += W_tile @ Fk_jtile  (B[k][c] = Fk[j0+k][c], from LDS)
    #pragma unroll
    for (int s = 0; s < 4; ++s) {
      const int kl = (4 * s + 2 * h) * OUT_C + r;
      const v2f bf = mk2(ldsFk[cur][kl], ldsFk[cur][kl + OUT_C]);
      acc = __builtin_amdgcn_wmma_f32_16x16x4_f32(false, aw[s], false, bf,
                                                  (short)0, acc, false, false);
    }
  }

  // Stage partials: numer in C/D layout -> row (i0 + v + 8h), col r
  #pragma unroll
  for (int v = 0; v < 8; ++v)
    unsafeAtomicAdd(&numer[(i0 + 8 * h + v) * OUT_C + r], acc[v]);
  unsafeAtomicAdd(&qsum[gi], qacc);
}

// ---------------------------------------------------------------------------
// Finalize: Fk_next = (numer + LAM*F0) / (qsum/D + LAM)
// ---------------------------------------------------------------------------
__global__ __launch_bounds__(256) void rung_finalize_kernel(
    const float* __restrict__ numer, const float* __restrict__ qsum,
    const float* __restrict__ Dd, const float* __restrict__ F0,
    float* __restrict__ out, int n) {
  const int t = blockIdx.x * blockDim.x + threadIdx.x;
  if (t >= n * OUT_C) return;
  const int i = t >> 4;
  const float q = qsum[i] / Dd[i] + LAM_C;
  out[t] = (numer[t] + LAM_C * F0[t]) / q;
}

// ---------------------------------------------------------------------------
extern "C" void kernel_launch(void* const* d_in, const int* in_sizes, int n_in,
                              void* d_out, int out_size, void* d_ws, size_t ws_size,
                              hipStream_t stream) {
  const float* A        = (const float*)d_in[0];
  const float* F        = (const float*)d_in[1];
  const float* W1       = (const float*)d_in[2];
  const float* b1       = (const float*)d_in[3];
  const float* W2       = (const float*)d_in[4];
  const float* b2       = (const float*)d_in[5];
  const float* log_lams = (const float*)d_in[6];

  const int n = in_sizes[1] / IN_DIM;  // 3072

  float* ws    = (float*)d_ws;
  float* F0    = ws;               // n*16
  float* fkA   = F0 + n * OUT_C;   // n*16
  float* fkB   = fkA + n * OUT_C;  // n*16
  float* Fn    = fkB + n * OUT_C;  // n*16
  float* numer = Fn + n * OUT_C;   // n*16
  float* sqv   = numer + n * OUT_C;// n
  float* qsum  = sqv + n;          // n
  float* Dd    = qsum + n;         // n
  float* Dsq   = Dd + n;           // n
  float* Dinv  = Dsq + n;          // n

  rung_mlp_kernel<<<n, 256, 0, stream>>>(F, W1, b1, W2, b2, F0, n);
  rung_degree_kernel<<<n, 256, 0, stream>>>(A, Dd, Dsq, Dinv, n);

  const int iTiles = n / 16;                    // 192
  const int jChunks = 8;
  const int tilesPerChunk = iTiles / jChunks;   // 24

  const float* fkin = F0;
  for (int k = 0; k < K_STEPS; ++k) {
    rung_prep_kernel<<<(n + 255) / 256, 256, 0, stream>>>(fkin, Dsq, Fn, sqv,
                                                          numer, qsum, n);
    dim3 grid(iTiles / WPB, jChunks);
    rung_step_kernel<<<grid, 256, 0, stream>>>(A, Fn, fkin, sqv, Dinv, log_lams,
                                               numer, qsum, k, n, tilesPerChunk);
    float* out = (k == K_STEPS - 1) ? (float*)d_out : ((k & 1) ? fkB : fkA);
    rung_finalize_kernel<<<(n * OUT_C + 255) / 256, 256, 0, stream>>>(
        numer, qsum, Dd, F0, out, n);
    fkin = out;
  }
}